// SimpleGAT_5179730559432
// MI455X (gfx1250) — compile-verified
//
#include <hip/hip_runtime.h>
#include <hip/hip_bf16.h>

// SimpleGAT on MI455X (gfx1250):
//   h_n = neighbor @ fc_w.T            (bias deferred; cancels in softmax)
//   alpha_n = h_n . attn[128:]         (h_c . attn[:128] cancels in softmax)
//   w = softmax(alpha); out = sum_n w_n h_n + fc_b
// Main GEMM uses V_WMMA_F32_16X16X4_F32 (exact f32, CDNA5 matrix path).

#define IN_DIM  128
#define OUT_DIM 128
#define NNB      64

typedef __attribute__((ext_vector_type(2))) float v2f;
typedef __attribute__((ext_vector_type(8))) float v8f;

#define XS_STRIDE 132   // 128 + 4: conflict-free 8B lane loads (132*idx + 2*half)
#define WK_STRIDE  36   // 32 + 4:  conflict-free 8B lane loads (36*idx + 2*half)

__global__ __launch_bounds__(256) void gat_wmma_kernel(
    const float* __restrict__ neighbor,  // [B, 64, 128]
    const float* __restrict__ fc_w,      // [128, 128] row-major (out, in)
    const float* __restrict__ fc_b,      // [128]
    const float* __restrict__ attn,      // [256]
    float* __restrict__ out,             // [B, 128]
    int B)
{
    __shared__ float Xs[NNB * XS_STRIDE];       // neighbor tile; reused for H
    __shared__ float Wk[OUT_DIM * WK_STRIDE];   // 32-wide K slice of fc_w
    __shared__ float An[OUT_DIM];               // attn[128:256]
    __shared__ float Al[NNB];
    __shared__ float Es[NNB];

    const int tid   = threadIdx.x;
    const int b     = blockIdx.x;
    const int wave  = tid >> 5;
    const int lane  = tid & 31;
    const int half  = lane >> 4;          // lane group 0: lanes 0-15, 1: 16-31
    const int idx   = lane & 15;
    const int mtile = wave & 3;           // 16-row tile over n (0..3)
    const int nbase = (wave >> 2) * 64;   // 64-col block over o (0 or 64)

    // ---- Stage neighbor[b] (64x128) into padded LDS, fully coalesced ----
    const float* nb = neighbor + (size_t)b * (NNB * IN_DIM);
#pragma unroll
    for (int v = 0; v < 8; ++v) {
        int f = (v * 256 + tid) * 4;      // flat float index, 4-aligned
        int n = f >> 7, i = f & 127;
        float4 d = *(const float4*)(nb + f);
        float* dst = &Xs[n * XS_STRIDE + i];
        dst[0] = d.x; dst[1] = d.y; dst[2] = d.z; dst[3] = d.w;
    }
    if (tid < OUT_DIM) An[tid] = attn[OUT_DIM + tid];

    v8f acc0 = {}, acc1 = {}, acc2 = {}, acc3 = {};

    // ---- GEMM: (64x128) @ fc_w.T, K staged in four 32-wide slices ----
    for (int kc = 0; kc < 4; ++kc) {
#pragma unroll
        for (int v = 0; v < 4; ++v) {
            int f = (v * 256 + tid) * 4;
            int o = f >> 5, kk = f & 31;
            float4 d = *(const float4*)(fc_w + o * IN_DIM + kc * 32 + kk);
            float* dst = &Wk[o * WK_STRIDE + kk];
            dst[0] = d.x; dst[1] = d.y; dst[2] = d.z; dst[3] = d.w;
        }
        __syncthreads();   // Wk ready (and Xs ready on first iteration)

#pragma unroll
        for (int k8 = 0; k8 < 8; ++k8) {
            // A 16x4 f32 layout: VGPR0 = K=2*half, VGPR1 = K=2*half+1; M = idx
            int ko = k8 * 4 + 2 * half;
            v2f a  = *(const v2f*)&Xs[(mtile * 16 + idx) * XS_STRIDE + kc * 32 + ko];
            // B 4x16 f32 layout (mirrored): b.x = W[o][k], b.y = W[o][k+1]; N = idx
            v2f b0 = *(const v2f*)&Wk[(nbase +  0 + idx) * WK_STRIDE + ko];
            v2f b1 = *(const v2f*)&Wk[(nbase + 16 + idx) * WK_STRIDE + ko];
            v2f b2 = *(const v2f*)&Wk[(nbase + 32 + idx) * WK_STRIDE + ko];
            v2f b3 = *(const v2f*)&Wk[(nbase + 48 + idx) * WK_STRIDE + ko];
            acc0 = __builtin_amdgcn_wmma_f32_16x16x4_f32(false, a, false, b0, (short)0, acc0, false, false);
            acc1 = __builtin_amdgcn_wmma_f32_16x16x4_f32(false, a, false, b1, (short)0, acc1, false, false);
            acc2 = __builtin_amdgcn_wmma_f32_16x16x4_f32(false, a, false, b2, (short)0, acc2, false, false);
            acc3 = __builtin_amdgcn_wmma_f32_16x16x4_f32(false, a, false, b3, (short)0, acc3, false, false);
        }
        __syncthreads();   // all reads of Wk (and, last iter, Xs) done
    }

    // ---- Spill H (no bias) into Xs (safe: barrier above) ----
    // D layout: VGPR r -> row r (lanes 0-15) / row r+8 (lanes 16-31), col = idx
#pragma unroll
    for (int r = 0; r < 8; ++r) {
        int m = mtile * 16 + r + 8 * half;
        Xs[m * XS_STRIDE + nbase +  0 + idx] = acc0[r];
        Xs[m * XS_STRIDE + nbase + 16 + idx] = acc1[r];
        Xs[m * XS_STRIDE + nbase + 32 + idx] = acc2[r];
        Xs[m * XS_STRIDE + nbase + 48 + idx] = acc3[r];
    }
    __syncthreads();

    // ---- alpha_n = H[n] . a_n  (h_c and bias terms cancel in softmax) ----
    if (tid < NNB) {
        float al = 0.f;
        for (int o = 0; o < OUT_DIM; ++o) al += Xs[tid * XS_STRIDE + o] * An[o];
        Al[tid] = al;
    }
    __syncthreads();

    // ---- softmax numerator ----
    if (tid < NNB) {
        float m = Al[0];
        for (int n = 1; n < NNB; ++n) m = fmaxf(m, Al[n]);   // LDS broadcast reads
        Es[tid] = __expf(Al[tid] - m);
    }
    __syncthreads();

    // ---- out[o] = (sum_n e_n * H[n][o]) / sum_n e_n + fc_b[o] ----
    if (tid < OUT_DIM) {
        float s = 0.f;
        for (int n = 0; n < NNB; ++n) s += Es[n];
        float v = 0.f;
        for (int n = 0; n < NNB; ++n) v += Es[n] * Xs[n * XS_STRIDE + tid];
        out[(size_t)b * OUT_DIM + tid] = v / s + fc_b[tid];
    }
}

extern "C" void kernel_launch(void* const* d_in, const int* in_sizes, int n_in,
                              void* d_out, int out_size, void* d_ws, size_t ws_size,
                              hipStream_t stream) {
    // setup_inputs order: center_feat, neighbor_feats, fc_w, fc_b, attn (all f32)
    const float* center   = (const float*)d_in[0]; (void)center; // cancels in softmax
    const float* neighbor = (const float*)d_in[1];
    const float* fc_w     = (const float*)d_in[2];
    const float* fc_b     = (const float*)d_in[3];
    const float* attn     = (const float*)d_in[4];
    float* outp = (float*)d_out;

    int B = in_sizes[0] / IN_DIM;   // 4096
    gat_wmma_kernel<<<B, 256, 0, stream>>>(neighbor, fc_w, fc_b, attn, outp, B);
}